// NLSFSPA_4690104287735
// MI455X (gfx1250) — compile-verified
//
#include <hip/hip_runtime.h>
#include <hip/hip_bf16.h>

// ---- problem constants (from reference setup) ----
constexpr int NB  = 16;      // batch
constexpr int C   = 64;      // channels
constexpr int HW  = 256*256; // spatial
constexpr int L   = 900;     // EV*EV selected positions
constexpr int LP  = 912;     // L padded to multiple of 16 (57 tiles)
constexpr int NT  = LP/16;   // 57 tiles along L
constexpr int SROW= 920;     // padded LDS row stride for score strip

typedef float v2f __attribute__((ext_vector_type(2)));
typedef float v8f __attribute__((ext_vector_type(8)));

__device__ inline v8f wmma_f32_k4(v2f a, v2f b, v8f c) {
  // D = A(16x4) * B(4x16) + C(16x16), all fp32 (V_WMMA_F32_16X16X4_F32)
  return __builtin_amdgcn_wmma_f32_16x16x4_f32(false, a, false, b, (short)0, c,
                                               false, false);
}

// ---------------------------------------------------------------------------
// K1: copy input -> out, and count channels >= 0 per spatial position.
// Fully coalesced streaming pass (this is the bandwidth-dominant kernel).
// ---------------------------------------------------------------------------
__global__ __launch_bounds__(256) void count_copy_kernel(const float* __restrict__ x,
                                                         float* __restrict__ out,
                                                         int* __restrict__ z) {
  const int pos = blockIdx.x * 256 + threadIdx.x;     // position within HW
  const int n   = blockIdx.y;
  const float* xn = x   + (size_t)n * C * HW + pos;
  float*       on = out + (size_t)n * C * HW + pos;
  int cnt = 0;
#pragma unroll 8
  for (int c = 0; c < C; ++c) {
    float v = xn[(size_t)c * HW];
    on[(size_t)c * HW] = v;
    cnt += (v >= 0.0f) ? 1 : 0;
  }
  z[n * HW + pos] = cnt;
}

// ---------------------------------------------------------------------------
// K2: per-batch stable selection of the L positions with smallest z.
// z in [0,64] -> 65-bin counting sort; within a bin positions are emitted in
// increasing index order (chunked scan + wave32 ballot prefix), which exactly
// reproduces jnp.argsort (stable) tie-breaking.
// ---------------------------------------------------------------------------
__global__ __launch_bounds__(256) void select_kernel(const int* __restrict__ z,
                                                     int* __restrict__ idx) {
  const int n = blockIdx.x;
  const int* zn = z + n * HW;
  __shared__ int hist[65];
  __shared__ int base[65];
  __shared__ int waveTot[8];
  __shared__ int run;
  const int t = threadIdx.x;
  const int lane = t & 31, w = t >> 5;

  if (t < 65) hist[t] = 0;
  __syncthreads();
  for (int p = t; p < HW; p += 256) atomicAdd(&hist[zn[p]], 1);
  __syncthreads();
  if (t == 0) {
    int acc = 0;
    for (int v = 0; v < 65; ++v) { base[v] = acc; acc += hist[v]; }
  }
  __syncthreads();

  for (int v = 0; v < 65; ++v) {
    if (base[v] >= L) break;        // all L slots already owned by smaller values
    if (hist[v] == 0) continue;
    if (t == 0) run = 0;
    __syncthreads();
    for (int chunk = 0; chunk < HW; chunk += 256) {
      const int p = chunk + t;
      const bool sel = (zn[p] == v);
      unsigned long long m = __ballot(sel);             // wave32: low 32 bits
      const int laneRank = __popcll(m & ((1ull << lane) - 1ull));
      if (lane == 0) waveTot[w] = __popcll(m);
      __syncthreads();
      int waveOff = 0;
      for (int i = 0; i < w; ++i) waveOff += waveTot[i];
      if (sel) {
        const int slot = base[v] + run + waveOff + laneRank;
        if (slot < L) idx[n * L + slot] = p;
      }
      __syncthreads();
      if (t == 0) {
        int tot = 0;
        for (int i = 0; i < 8; ++i) tot += waveTot[i];
        run += tot;
      }
      __syncthreads();
      if (run >= hist[v] || base[v] + run >= L) break;  // uniform (LDS after barrier)
    }
    __syncthreads();
  }
}

// ---------------------------------------------------------------------------
// K3: gather qT[n][l][c] = flat[n][c][idx[n][l]]  (K-contiguous layout so the
// projection's WMMA B-fragments are single b64 loads; pad rows zeroed)
// ---------------------------------------------------------------------------
__global__ __launch_bounds__(256) void gather_kernel(const float* __restrict__ x,
                                                     const int* __restrict__ idx,
                                                     float* __restrict__ qT) {
  const int gid = blockIdx.x * 256 + threadIdx.x;
  if (gid >= NB * LP * C) return;
  const int c = gid % C;
  const int l = (gid / C) % LP;
  const int n = gid / (C * LP);
  float v = 0.0f;
  if (l < L) {
    const int p = idx[n * L + l];
    v = x[(size_t)n * C * HW + (size_t)c * HW + p];
  }
  qT[gid] = v;
}

// ---------------------------------------------------------------------------
// K4: projections [e1;e2;ea] = [W1;W2;Wa] * q  (+bias, PReLU).
//   rows 0..63  -> e12T[n][l][64]  (cols 0..31 = e1, 32..63 = e2, K-contiguous)
//   rows 64..127-> ea [n][64][LP]  (m-contiguous for the attn A-fragments)
// One wave per 16x16 tile, K=64 via 16 chained v_wmma_f32_16x16x4_f32.
// ---------------------------------------------------------------------------
__global__ __launch_bounds__(32) void proj_kernel(const float* __restrict__ qT,
                                                  const float* __restrict__ W1,
                                                  const float* __restrict__ b1,
                                                  const float* __restrict__ a1,
                                                  const float* __restrict__ W2,
                                                  const float* __restrict__ b2,
                                                  const float* __restrict__ a2,
                                                  const float* __restrict__ Wa,
                                                  const float* __restrict__ ba,
                                                  const float* __restrict__ aa,
                                                  float* __restrict__ e12T,
                                                  float* __restrict__ ea) {
  const int tileN = blockIdx.x;       // 0..56  (L tiles)
  const int tileM = blockIdx.y;       // 0..7   (128 output rows / 16)
  const int n     = blockIdx.z;
  const int lane  = threadIdx.x;
  const int rowA  = tileM * 16 + (lane & 15);   // A-fragment row (M)
  const int colN  = tileN * 16 + (lane & 15);   // B/D column (N = l)
  const int khi   = (lane >> 4) * 2;            // K-pair select (ISA A layout)

  const float* wrow;
  if      (rowA < 32) wrow = W1 + (size_t)rowA * 64;
  else if (rowA < 64) wrow = W2 + (size_t)(rowA - 32) * 64;
  else                wrow = Wa + (size_t)(rowA - 64) * 64;

  const float* qn = qT + (size_t)n * LP * C;    // [l][64]
  v8f acc = {};
#pragma unroll
  for (int k0 = 0; k0 < 64; k0 += 4) {
    v2f a = *(const v2f*)(wrow + k0 + khi);                       // b64
    v2f b = *(const v2f*)(qn + (size_t)colN * C + k0 + khi);      // b64
    acc = wmma_f32_k4(a, b, acc);
  }

  const int mh = (lane >> 4) * 8;
  const int M0 = tileM * 16 + mh;               // first D row for this lane
  float vo[8];
#pragma unroll
  for (int i = 0; i < 8; ++i) {
    const int M = M0 + i;
    float bi, sl;
    if      (M < 32) { bi = b1[M];       sl = a1[0]; }
    else if (M < 64) { bi = b2[M - 32];  sl = a2[0]; }
    else             { bi = ba[M - 64];  sl = aa[0]; }
    float v = acc[i] + bi;
    vo[i] = (v >= 0.0f) ? v : sl * v;           // PReLU
  }

  if (tileM < 4) {
    // e12T[n][colN][M0..M0+7]  -> two 16B stores (M0 multiple of 8)
    float* dst = e12T + (size_t)n * LP * 64 + (size_t)colN * 64 + M0;
    *(float4*)(dst + 0) = make_float4(vo[0], vo[1], vo[2], vo[3]);
    *(float4*)(dst + 4) = make_float4(vo[4], vo[5], vo[6], vo[7]);
  } else {
    float* en = ea + (size_t)n * 64 * LP;
#pragma unroll
    for (int i = 0; i < 8; ++i)
      en[(size_t)(M0 - 64 + i) * LP + colN] = vo[i];
  }
}

// ---------------------------------------------------------------------------
// K5: attention. One workgroup per (l-tile, batch):
//  pre:    async-stage the 16x32 e1 l-strip into LDS (global_load_async_to_lds)
//  phase1: S[16][912] strip = e1^T * e2 (K=32) via WMMA (A hoisted to regs)
//  phase2: masked softmax over m < 900 (scratch reuses attnT buffer)
//  phase3: attn[64][16] = ea * S^T (K=912) via WMMA, waves split c-tiles and
//          K halves, partials combined with ds_add_f32 LDS atomics
//  phase4: residual add + scatter to out via idx
// ---------------------------------------------------------------------------
__global__ __launch_bounds__(256) void attn_kernel(const float* __restrict__ e12T,
                                                   const float* __restrict__ ea,
                                                   const float* __restrict__ qT,
                                                   const int* __restrict__ idx,
                                                   float* __restrict__ out) {
  const int ltile = blockIdx.x;       // 0..56
  const int n     = blockIdx.y;
  __shared__ float S[16][SROW];       // score strip (l-tile rows x all m)
  __shared__ float attnT[64][16];     // attention tile (also softmax scratch)
  __shared__ float aStage[16][32];    // e1 strip for this l-tile

  const float* e12n = e12T + (size_t)n * LP * 64;   // [l][64]
  const float* ean  = ea   + (size_t)n * 64 * LP;   // [c][LP]

  const int t = threadIdx.x, lane = t & 31, w = t >> 5;
  const int lbase = ltile * 16;
  const int khi = (lane >> 4) * 2;

  // ---- async stage of the e1 strip (16 rows x 32 cols = 2KB) ----
  if (t < 128) {
    const float* gsrc = e12n + (size_t)(lbase + (t >> 3)) * 64 + (t & 7) * 4;
    unsigned lofs = (unsigned)(uintptr_t)(&aStage[0][0]) + (unsigned)t * 16u;
    unsigned long long ga = (unsigned long long)(uintptr_t)gsrc;
    asm volatile("global_load_async_to_lds_b128 %0, %1, off"
                 :: "v"(lofs), "v"(ga) : "memory");
  }
  asm volatile("s_wait_asynccnt 0" ::: "memory");
  __syncthreads();

  // A fragments are invariant across all m-tiles: hoist to registers.
  v2f aR[8];
#pragma unroll
  for (int ks = 0; ks < 8; ++ks)
    aR[ks] = *(const v2f*)(&aStage[lane & 15][ks * 4 + khi]);   // ds_load_b64

  // ---- phase 1: score strip ----
  for (int mt = w; mt < NT; mt += 8) {
    const int mbase = mt * 16;
    const float* e2row = e12n + (size_t)(mbase + (lane & 15)) * 64 + 32;
    v8f acc = {};
#pragma unroll
    for (int ks = 0; ks < 8; ++ks) {
      v2f b = *(const v2f*)(e2row + ks * 4 + khi);              // b64
      acc = wmma_f32_k4(aR[ks], b, acc);
    }
    const int mh = (lane >> 4) * 8;
#pragma unroll
    for (int i = 0; i < 8; ++i)
      S[mh + i][mbase + (lane & 15)] = acc[i];
  }
  __syncthreads();

  // ---- phase 2: softmax over m (mask m >= 900); scratch aliases attnT ----
  float (*red)[17] = (float (*)[17])(&attnT[0][0]);
  const int r = t >> 4, j = t & 15;   // 16 threads per row
  float mx = -3.4e38f;
  for (int m = j; m < L; m += 16) mx = fmaxf(mx, S[r][m]);
  red[r][j] = mx;
  __syncthreads();
  if (j == 0) {
    float m2 = red[r][0];
    for (int i = 1; i < 16; ++i) m2 = fmaxf(m2, red[r][i]);
    red[r][16] = m2;
  }
  __syncthreads();
  const float rowmax = red[r][16];
  __syncthreads();
  float sm = 0.0f;
  for (int m = j; m < LP; m += 16) {
    float p = 0.0f;
    if (m < L) { p = __expf(S[r][m] - rowmax); sm += p; }
    S[r][m] = p;                      // padded columns forced to 0
  }
  red[r][j] = sm;
  __syncthreads();
  if (j == 0) {
    float s2 = 0.0f;
    for (int i = 0; i < 16; ++i) s2 += red[r][i];
    red[r][16] = 1.0f / s2;
  }
  __syncthreads();
  const float inv = red[r][16];
  for (int m = j; m < L; m += 16) S[r][m] *= inv;
  __syncthreads();                    // all reads of softmax scratch done
  for (int i = t; i < 64 * 16; i += 256) (&attnT[0][0])[i] = 0.0f;
  __syncthreads();

  // ---- phase 3: attn = ea * S^T  (waves: 4 c-tiles x 2 K halves) ----
  {
    const int ctile = w & 3;
    const int kh    = w >> 2;
    const float* earow = ean + (size_t)(ctile * 16 + (lane & 15)) * LP;
    v8f acc = {};
#pragma unroll 2
    for (int k0 = kh * 456; k0 < kh * 456 + 456; k0 += 4) {
      v2f a = *(const v2f*)(earow + k0 + khi);                  // b64
      v2f b = *(const v2f*)(&S[lane & 15][k0 + khi]);           // ds_load_b64
      acc = wmma_f32_k4(a, b, acc);
    }
    const int mh = (lane >> 4) * 8;
#pragma unroll
    for (int i = 0; i < 8; ++i)
      atomicAdd(&attnT[ctile * 16 + mh + i][lane & 15], acc[i]);  // ds_add_f32
  }
  __syncthreads();

  // ---- phase 4: residual + scatter ----
  const float* qn = qT + (size_t)n * LP * C;     // [l][64]
  for (int i = t; i < 64 * 16; i += 256) {
    const int c  = i >> 4;
    const int ll = i & 15;
    const int lg = lbase + ll;
    if (lg < L) {
      const int p = idx[n * L + lg];
      out[(size_t)n * C * HW + (size_t)c * HW + p] =
          attnT[c][ll] + qn[(size_t)lg * C + c];
    }
  }
}

// ---------------------------------------------------------------------------
extern "C" void kernel_launch(void* const* d_in, const int* in_sizes, int n_in,
                              void* d_out, int out_size, void* d_ws, size_t ws_size,
                              hipStream_t stream) {
  (void)in_sizes; (void)n_in; (void)out_size; (void)ws_size;
  const float* x  = (const float*)d_in[0];
  const float* W1 = (const float*)d_in[1];
  const float* b1 = (const float*)d_in[2];
  const float* a1 = (const float*)d_in[3];
  const float* W2 = (const float*)d_in[4];
  const float* b2 = (const float*)d_in[5];
  const float* a2 = (const float*)d_in[6];
  const float* Wa = (const float*)d_in[7];
  const float* ba = (const float*)d_in[8];
  const float* aa = (const float*)d_in[9];
  float* out = (float*)d_out;

  // workspace layout (256B aligned slices)
  char* ws = (char*)d_ws;
  auto align256 = [](size_t v) { return (v + 255) & ~(size_t)255; };
  size_t off = 0;
  int*   z    = (int*)(ws + off);   off += align256((size_t)NB * HW * sizeof(int));
  int*   idx  = (int*)(ws + off);   off += align256((size_t)NB * L * sizeof(int));
  float* qT   = (float*)(ws + off); off += align256((size_t)NB * LP * C * sizeof(float));
  float* e12T = (float*)(ws + off); off += align256((size_t)NB * LP * 64 * sizeof(float));
  float* ea   = (float*)(ws + off); off += align256((size_t)NB * 64 * LP * sizeof(float));

  // K1: stream copy + channel-positive counts (bandwidth-dominant)
  count_copy_kernel<<<dim3(HW / 256, NB), 256, 0, stream>>>(x, out, z);
  // K2: stable top-L selection (counting sort over 65 bins)
  select_kernel<<<NB, 256, 0, stream>>>(z, idx);
  // K3: gather qT (K-contiguous)
  gather_kernel<<<(NB * LP * C) / 256, 256, 0, stream>>>(x, idx, qT);
  // K4: projections via fp32 WMMA
  proj_kernel<<<dim3(NT, 8, NB), 32, 0, stream>>>(qT, W1, b1, a1, W2, b2, a2,
                                                  Wa, ba, aa, e12T, ea);
  // K5: attention (async e1 stage + score WMMA + softmax + attn WMMA + scatter)
  attn_kernel<<<dim3(NT, NB), 256, 0, stream>>>(e12T, ea, qT, idx, out);
}